// TempestRegridder_2310692405549
// MI455X (gfx1250) — compile-verified
//
#include <hip/hip_runtime.h>

// ---------------------------------------------------------------------------
// TempestRegridder: y[ch, rows[k]] += S[k] * x[ch, cols[k]],  ch in [0,64)
//   x : [B=4, C=16, N_IN=721*1440]  -> 64 planes of N_IN floats
//   y : [B=4, C=16, N_OUT=361*720]  -> 64 planes of N_OUT floats (d_out)
// Memory/atomic bound (~0.3 FLOP/B). Scale-multiply runs on the matrix unit:
// per 16-nnz x 16-ch tile, D = sum_g A_g(16x4) * B_g(4x16) with shifted-
// diagonal B_g carrying S[k]  =>  D[ch,t] = S[t]*x[ch,cols[t]].
// ---------------------------------------------------------------------------

typedef __attribute__((ext_vector_type(2))) float v2f;
typedef __attribute__((ext_vector_type(8))) float v8f;

#define H_IN   721
#define W_IN   1440
#define H_OUT  361
#define W_OUT  720
#define N_IN   (H_IN * W_IN)      // 1038240
#define N_OUT  (H_OUT * W_OUT)    // 259920
#define NNZ    1048576
#define NCHUNK (NNZ / 16)         // 65536, exact

__global__ __launch_bounds__(256)
void regrid_zero_kernel(float* __restrict__ y, int n) {
    int i = blockIdx.x * blockDim.x + threadIdx.x;
    int stride = gridDim.x * blockDim.x;
    for (; i < n; i += stride) y[i] = 0.0f;
}

__global__ __launch_bounds__(256)
void regrid_coo_wmma_kernel(const float* __restrict__ x,
                            const float* __restrict__ S,
                            const int*   __restrict__ rows,
                            const int*   __restrict__ cols,
                            float*       __restrict__ y) {
    const int lane = threadIdx.x & 31;
    const int n    = lane & 15;   // A: channel row; B/D: nnz slot within tile
    const int half = lane >> 4;   // selects K placement (K = 2*half + v)

    // Wave id is uniform; make it provably scalar so the chunk loop is a
    // uniform s_cbranch loop (no exec-mask juggling, EXEC stays all-ones).
    const int gw = __builtin_amdgcn_readfirstlane(
        blockIdx.x * (blockDim.x >> 5) + (threadIdx.x >> 5));
    const int totalWaves = gridDim.x * (blockDim.x >> 5);

    const int chgroup    = gw & 3;            // which 16 channels
    const int stream     = gw >> 2;           // which nnz stream
    const int numStreams = totalWaves >> 2;
    const int chbase     = chgroup * 16;

    // Per-lane x plane base: lane handles channel chbase+n on the A side.
    const float* __restrict__ xp = x + (size_t)(chbase + n) * N_IN;

    for (int chunk = stream; chunk < NCHUNK; chunk += numStreams) {
        const int kbase = chunk * 16;
        const int kb2   = kbase + 2 * half;   // lane's K-slot base

        // Phase 1: batch all index/weight loads (one clause, immediate offsets).
        // Lane's nonzero ids: kbase + 4g + (2*half + v)  ->  kb2 + 4g + v.
        int   colv[8];
        float sv[8];
        #pragma unroll
        for (int i = 0; i < 8; ++i) {         // i = 2g+v
            const int off = ((i >> 1) << 2) | (i & 1);   // 0,1,4,5,8,9,12,13
            colv[i] = cols[kb2 + off];
            sv[i]   = S[kb2 + off];           // unconditional: no exec branches
        }
        const int r = rows[kbase + n];        // scatter row for slot t = n

        // Phase 2: batch the 8 x-gathers (independent once cols arrive).
        float av[8];
        #pragma unroll
        for (int i = 0; i < 8; ++i) av[i] = xp[colv[i]];

        // Phase 3: 4 accumulated WMMAs. B_g is a shifted diagonal: the S
        // value survives only in the lane/vgpr where column n == 4g+K.
        v8f d = {};
        #pragma unroll
        for (int g = 0; g < 4; ++g) {
            v2f a, b;
            a[0] = av[2 * g + 0];
            a[1] = av[2 * g + 1];
            b[0] = (n == 4 * g + 2 * half + 0) ? sv[2 * g + 0] : 0.0f;
            b[1] = (n == 4 * g + 2 * half + 1) ? sv[2 * g + 1] : 0.0f;
            d = __builtin_amdgcn_wmma_f32_16x16x4_f32(
                    false, a, false, b, (short)0, d, false, false);
        }

        // Phase 4: scatter. Lane owns nnz slot t = n; VGPR v holds channel
        // M = v + 8*half  ->  8 atomics at constant plane offsets (one clause).
        float* yb = y + (size_t)(chbase + 8 * half) * N_OUT + (size_t)r;
        #pragma unroll
        for (int v = 0; v < 8; ++v) {
            atomicAdd(yb + (size_t)v * N_OUT, d[v]);
        }
    }
}

extern "C" void kernel_launch(void* const* d_in, const int* in_sizes, int n_in,
                              void* d_out, int out_size, void* d_ws, size_t ws_size,
                              hipStream_t stream) {
    (void)in_sizes; (void)n_in; (void)d_ws; (void)ws_size;

    const float* x    = (const float*)d_in[0];
    const float* S    = (const float*)d_in[1];
    const int*   rows = (const int*)d_in[2];
    const int*   cols = (const int*)d_in[3];
    float*       y    = (float*)d_out;

    // d_out is poisoned by the harness; atomics accumulate, so zero it first.
    regrid_zero_kernel<<<4096, 256, 0, stream>>>(y, out_size);

    // 2048 blocks * 8 waves = 16384 waves = 4096 nnz-streams x 4 ch-groups;
    // each stream handles 65536/4096 = 16 chunks of 16 nonzeros.
    regrid_coo_wmma_kernel<<<2048, 256, 0, stream>>>(x, S, rows, cols, y);
}